// RenderingLayer_57501022159347
// MI455X (gfx1250) — compile-verified
//
#include <hip/hip_runtime.h>
#include <cmath>
#include <cstdint>

#define MAXSG 32
#define PI_F 3.14159265358979323846f
#define INV_PI_F 0.3183098861837907f
#define EPS_F 1e-7f

typedef unsigned int u32x4 __attribute__((ext_vector_type(4)));
typedef int i32x8 __attribute__((ext_vector_type(8)));
typedef int i32x4 __attribute__((ext_vector_type(4)));

__device__ __forceinline__ float satf(float x) { return fminf(fmaxf(x, 0.0f), 1.0f); }
__device__ __forceinline__ float clampf(float x, float lo, float hi) { return fminf(fmaxf(x, lo), hi); }

__device__ __forceinline__ float srgb_lin(float x) {
  // ((max(x,0.04045)+0.055)/1.055)^2.4  vs  x/12.92
  float p = __powf((fmaxf(x, 0.04045f) + 0.055f) * (1.0f / 1.055f), 2.4f);
  return (x >= 0.04045f) ? p : x * (1.0f / 12.92f);
}

__device__ __forceinline__ float ggx1(float a2, float ndx) {
  // 1 / (ndx + safe_sqrt(a2 + (1-a2) ndx^2)); denominator > 0 always
  return 1.0f / (ndx + sqrtf(fmaxf(a2 + (1.0f - a2) * ndx * ndx, EPS_F)));
}

__global__ __launch_bounds__(256) void pbr_sg_render(
    const float* __restrict__ diffuse, const float* __restrict__ specular,
    const float* __restrict__ roughness, const float* __restrict__ normal,
    const float* __restrict__ depth, const float* __restrict__ mask,
    const float* __restrict__ camera_pos, const float* __restrict__ light_pos,
    const float* __restrict__ light_color, const float* __restrict__ sgs,
    float* __restrict__ out, int H, int W, int S)
{
  // 3D grid: x tiles a row (256 px), y = row, z = batch  -> no integer div/mod anywhere
  const int b   = (int)blockIdx.z;
  const int i   = (int)blockIdx.y;
  const int tid = (int)threadIdx.x;
  const int j   = (int)blockIdx.x * (int)blockDim.x + tid;
  const int HW  = H * W;

  __shared__ __align__(16) float sRaw[7 * MAXSG];
  __shared__ float4 sSG[MAXSG][4];
  __shared__ float sIx[MAXSG];

  const float* gsg = sgs + (size_t)b * 7 * (size_t)S;
  const int nRaw = 7 * S;

  // ---- Stage the per-batch SG table (7*S floats) into LDS via the TDM ----
#if __has_builtin(__builtin_amdgcn_tensor_load_to_lds)
  if (tid < 32) {  // wave 0 issues the tensor DMA (EXEC ignored by TDM)
    uint32_t ldsAddr = (uint32_t)(uintptr_t)&sRaw[0];   // low 32 bits of generic shared ptr = LDS offset
    uint64_t ga = (uint64_t)(uintptr_t)gsg;
    // D# group0: count=1 (valid), lds_addr, global_addr[56:0], type=2 ("image")
    u32x4 g0 = { 1u, ldsAddr, (uint32_t)ga,
                 ((uint32_t)((ga >> 32) & 0x1FFFFFFull)) | 0x80000000u };
    // D# group1: data_size=4B; tensor_dim0=nRaw; tensor_dim1=1; tile_dim0=nRaw;
    //            tile_dim1=1; tile_dim2=0; tensor_dim0_stride=nRaw; workgroup_mask=0
    i32x8 g1 = { (int)(2u << 16),
                 (int)(((uint32_t)nRaw & 0xFFFFu) << 16),
                 (int)(1u << 16),
                 (int)(((uint32_t)nRaw & 0xFFFFu) << 16),
                 1,
                 nRaw,
                 0, 0 };
    i32x4 gz4 = { 0, 0, 0, 0 };
    i32x8 gz8 = { 0, 0, 0, 0, 0, 0, 0, 0 };
    __builtin_amdgcn_tensor_load_to_lds(g0, g1, gz4, gz4, gz8, 0);
#if __has_builtin(__builtin_amdgcn_s_wait_tensorcnt)
    __builtin_amdgcn_s_wait_tensorcnt(0);
#else
    asm volatile("s_wait_tensorcnt 0x0" ::: "memory");
#endif
  }
#else
  for (int k = tid; k < nRaw; k += (int)blockDim.x) sRaw[k] = gsg[k];
#endif
  __syncthreads();

  // ---- Per-(batch,SG) derived constants, computed once per block ----
  if (tid < S) {
    const int s = tid;
    float A0 = sRaw[0 * S + s], A1 = sRaw[1 * S + s], A2 = sRaw[2 * S + s];
    float ax = sRaw[3 * S + s], ay = sRaw[4 * S + s], az = sRaw[5 * S + s];
    float sh = sRaw[6 * S + s];
    float eml  = __expf(-sh);
    float em2l = eml * eml;
    float rl   = (sh == 0.0f) ? 0.0f : 1.0f / sh;           // div_no_nan(1, sharp)
    float scl  = 1.0f + 2.0f * em2l - rl;
    float bia  = (eml - em2l) * rl - em2l;
    float xs   = sqrtf(fmaxf(1.0f - scl, EPS_F));
    float ib   = 2.0f * PI_F * (1.0f - __expf(-2.0f * sh)) * rl;  // integral / amp
    sSG[s][0] = make_float4(ax, ay, az, (1.0f / (4.0f * 0.36f)) * xs);  // axis, x1=c1*x
    sSG[s][1] = make_float4(sh * ax, sh * ay, sh * az, sh);             // s2*axis, s2
    sSG[s][2] = make_float4(A0 * ib, A1 * ib, A2 * ib, scl);            // integral, scale
    sSG[s][3] = make_float4(A0, A1, A2, bia);                           // amp, bias
    sIx[s] = 1.0f / xs;
  }
  __syncthreads();

  if (j >= W) return;
  const int pix = i * W + j;

  const size_t p3 = (size_t)b * 3 * HW + pix;   // 3-channel plane base (+ c*HW)
  const size_t p1 = (size_t)b * HW + pix;       // 1-channel base

  // ---- loads (coalesced per plane) ----
  float df0 = diffuse[p3], df1 = diffuse[p3 + HW], df2 = diffuse[p3 + 2 * (size_t)HW];
  float sp0 = specular[p3], sp1 = specular[p3 + HW], sp2 = specular[p3 + 2 * (size_t)HW];
  float nr0 = normal[p3], nr1 = normal[p3 + HW], nr2 = normal[p3 + 2 * (size_t)HW];
  float rough = roughness[p1];
  float dep   = depth[p1];
  float msk   = mask[p1];
  float cpx = camera_pos[b * 3 + 0], cpy = camera_pos[b * 3 + 1], cpz = camera_pos[b * 3 + 2];
  float lpx = light_pos[b * 3 + 0],  lpy = light_pos[b * 3 + 1],  lpz = light_pos[b * 3 + 2];
  float lc0 = light_color[b * 3 + 0], lc1 = light_color[b * 3 + 1], lc2 = light_color[b * 3 + 2];

  // ---- perturbed mesh ----
  const float r = 0.40414518843273806f;            // 0.7 * tan(30 deg)
  const float step = 2.0f * r / (float)(H - 1);
  float rd  = 1.0f / (5.0f * dep + 0.7f);          // uncompress depth (denom >= 0.7)
  float pmx = (r - j * step) * rd;
  float pmy = (r - i * step) * rd;
  float pmz = rd;

  // ---- view / light / half vectors ----
  float vx = cpx - pmx, vy = cpy - pmy, vz = cpz - pmz;
  float vinv = rsqrtf(fmaxf(vx * vx + vy * vy + vz * vz, EPS_F));
  vx *= vinv; vy *= vinv; vz *= vinv;

  float lvx = lpx - pmx, lvy = lpy - pmy, lvz = lpz - pmz;
  float lm2 = lvx * lvx + lvy * lvy + lvz * lvz;
  float linv = rsqrtf(fmaxf(lm2, EPS_F));
  float lx = lvx * linv, ly = lvy * linv, lz = lvz * linv;

  float hx = lx + vx, hy = ly + vy, hz = lz + vz;
  float hinv = rsqrtf(fmaxf(hx * hx + hy * hy + hz * hz, EPS_F));
  hx *= hinv; hy *= hinv; hz *= hinv;

  // ---- normal: normalize(2n-1) * (-1, 1, -1) ----
  float n0 = nr0 * 2.0f - 1.0f, n1 = nr1 * 2.0f - 1.0f, n2 = nr2 * 2.0f - 1.0f;
  float ninv = rsqrtf(fmaxf(n0 * n0 + n1 * n1 + n2 * n2, EPS_F));
  float nx = -n0 * ninv, ny = n1 * ninv, nz = -n2 * ninv;

  float ndl = satf(nx * lx + ny * ly + nz * lz);
  float nv  = nx * vx + ny * vy + nz * vz;
  float ndv = clampf(nv, 1e-5f, 1.0f);
  float ndh = satf(nx * hx + ny * hy + nz * hz);
  float ldh = satf(lx * hx + ly * hy + lz * hz);
  float vdh = satf(vx * hx + vy * hy + vz * hz);

  float lrcp = (lm2 == 0.0f) ? 0.0f : 1.0f / lm2;  // div_no_nan(lc, |l_vec|^2)
  float li0 = lc0 * lrcp, li1 = lc1 * lrcp, li2 = lc2 * lrcp;

  float dl0 = srgb_lin(df0), dl1 = srgb_lin(df1), dl2 = srgb_lin(df2);
  float sl0 = srgb_lin(sp0), sl1 = srgb_lin(sp1), sl2 = srgb_lin(sp2);

  // ---- direct specular (GGX) + direct diffuse ----
  float alpha = clampf(rough * rough, 0.001f, 1.0f);
  float a2b = alpha * alpha;
  float ct = 1.0f - ldh, ct2 = ct * ct, f5 = ct2 * ct2 * ct;
  float F0 = sl0 + (1.0f - sl0) * f5;
  float F1 = sl1 + (1.0f - sl1) * f5;
  float F2 = sl2 + (1.0f - sl2) * f5;
  float G = (2.0f * ndl * ggx1(a2b, ndl)) * (2.0f * ndv * ggx1(a2b, ndv));
  float dn = ndh * ndh * (a2b - 1.0f) + 1.0f;
  float Dd = PI_F * dn * dn;
  float Dt = (Dd == 0.0f) ? 0.0f : a2b / Dd;
  float s4 = 4.0f * ndl;
  float sfac = (s4 == 0.0f) ? 0.0f : (G * Dt) / s4;
  if (ndh < EPS_F || ldh * ndl < EPS_F || vdh * ndv < EPS_F) sfac = 0.0f;
  float kd = INV_PI_F * ndv;
  float dir0 = (F0 * sfac + dl0 * kd * (1.0f - F0)) * li0;
  float dir1 = (F1 * sfac + dl1 * kd * (1.0f - F1)) * li1;
  float dir2 = (F2 * sfac + dl2 * kd * (1.0f - F2)) * li2;
  if (ndl < EPS_F) { dir0 = 0.0f; dir1 = 0.0f; dir2 = 0.0f; }

  // ---- env: per-pixel SG-independent factors ----
  float a2s   = clampf(rough * rough, 0.001f, 1.0f);
  float ampw  = 1.0f / (PI_F * a2s);
  float sharpw = (2.0f / a2s) / (4.0f * clampf(nv, 1e-4f, 1.0f));
  float wdx = 2.0f * nv * nx - vx, wdy = 2.0f * nv * ny - vy, wdz = 2.0f * nv * nz - vz;
  float ndl_s = satf(nx * wdx + ny * wdy + nz * wdz);
  float ndv_s = satf(nv);
  float h2x = wdx + vx, h2y = wdy + vy, h2z = wdz + vz;
  float h2i = rsqrtf(fmaxf(h2x * h2x + h2y * h2y + h2z * h2z, EPS_F));
  float ldh_s = satf((wdx * h2x + wdy * h2y + wdz * h2z) * h2i);
  float Gs = ggx1(a2s, ndl_s) * ggx1(a2s, ndv_s);
  float powt = exp2f((-5.55473f * ldh_s - 6.98316f) * ldh_s);
  float GK = Gs * ndl_s;
  float K0 = (sl0 + (1.0f - sl0) * powt) * GK;
  float K1 = (sl1 + (1.0f - sl1) * powt) * GK;
  float K2c = (sl2 + (1.0f - sl2) * powt) * GK;
  float pwx = sharpw * wdx, pwy = sharpw * wdy, pwz = sharpw * wdz;
  float K2pi = 2.0f * PI_F * ampw;
  float dp0 = dl0 * INV_PI_F, dp1 = dl1 * INV_PI_F, dp2 = dl2 * INV_PI_F;

  // ---- SG loop: 2 exps + 1 sqrt + ~25 VALU per iteration ----
  float aD0 = 0.0f, aD1 = 0.0f, aD2 = 0.0f;   // sum d_s * integral_s
  float aS0 = 0.0f, aS1 = 0.0f, aS2 = 0.0f;   // sum D_s * amp_s
#pragma unroll 4
  for (int s = 0; s < S; ++s) {
    float4 q0 = sSG[s][0];
    float4 q1 = sSG[s][1];
    float4 q2 = sSG[s][2];
    float4 q3 = sSG[s][3];
    // diffuse SG term
    float mudn = satf(q0.x * nx + q0.y * ny + q0.z * nz);
    float x0t = 0.36f * mudn;
    float nterm = x0t + q0.w;
    float y = (fabsf(x0t) <= q0.w) ? nterm * nterm * sIx[s] : mudn;
    float dcoef = q2.w * y + q3.w;
    aD0 += dcoef * q2.x; aD1 += dcoef * q2.y; aD2 += dcoef * q2.z;
    // specular SG inner product with warped lobe
    float tx = pwx + q1.x, ty = pwy + q1.y, tz = pwz + q1.z;
    float um = sqrtf(fmaxf(tx * tx + ty * ty + tz * tz, EPS_F));
    float e1 = __expf(um - sharpw - q1.w);
    float oth = 1.0f - __expf(-2.0f * um);
    float Ds = K2pi * e1 * oth / um;
    aS0 += Ds * q3.x; aS1 += Ds * q3.y; aS2 += Ds * q3.z;
  }

  float env0 = fmaxf(aS0 * K0, 0.0f) + aD0 * dp0;
  float env1 = fmaxf(aS1 * K1, 0.0f) + aD1 * dp1;
  float env2 = fmaxf(aS2 * K2c, 0.0f) + aD2 * dp2;

  out[p3]                     = dir0 + env0 * msk;
  out[p3 + HW]                = dir1 + env1 * msk;
  out[p3 + 2 * (size_t)HW]    = dir2 + env2 * msk;
}

extern "C" void kernel_launch(void* const* d_in, const int* in_sizes, int n_in,
                              void* d_out, int out_size, void* d_ws, size_t ws_size,
                              hipStream_t stream) {
  const float* diffuse   = (const float*)d_in[0];
  const float* specular  = (const float*)d_in[1];
  const float* roughness = (const float*)d_in[2];
  const float* normal    = (const float*)d_in[3];
  const float* depth     = (const float*)d_in[4];
  const float* mask      = (const float*)d_in[5];
  const float* cam       = (const float*)d_in[6];
  const float* lpos      = (const float*)d_in[7];
  const float* lcol      = (const float*)d_in[8];
  const float* sgs       = (const float*)d_in[9];
  float* out = (float*)d_out;

  int B = in_sizes[6] / 3;
  if (B <= 0) B = 1;
  long long HW = in_sizes[2] / B;                 // roughness is [B,1,H,W]
  int H = (int)(std::sqrt((double)HW) + 0.5);
  if (H <= 0) H = 1;
  int W = (int)(HW / H);
  int S = in_sizes[9] / (7 * B);
  if (S > MAXSG) S = MAXSG;

  const int threads = 256;
  dim3 grid((unsigned)((W + threads - 1) / threads), (unsigned)H, (unsigned)B);
  pbr_sg_render<<<grid, threads, 0, stream>>>(
      diffuse, specular, roughness, normal, depth, mask,
      cam, lpos, lcol, sgs, out, H, W, S);
}